// MessagePassing_2585570312451
// MI455X (gfx1250) — compile-verified
//
#include <hip/hip_runtime.h>
#include <stdint.h>

#define FEAT 16
#define TILE_EDGES 64          // edges per tile (256 thr = 4 lanes/edge)

typedef float v4f   __attribute__((ext_vector_type(4)));
typedef int   v2i32 __attribute__((vector_size(8)));     // b64 payload type

typedef __attribute__((address_space(1))) v2i32 gv2i;    // global int2
typedef __attribute__((address_space(3))) v2i32 lv2i;    // LDS int2

// ---- CDNA5 async-to-LDS helpers (ASYNCcnt path) ---------------------------
__device__ __forceinline__ void async_copy_b64(const long long* g, long long* l) {
    __builtin_amdgcn_global_load_async_to_lds_b64((gv2i*)g, (lv2i*)l,
                                                  /*offset*/0, /*cpol*/0);
}

template <int N>
__device__ __forceinline__ void wait_asynccnt() {
#if __has_builtin(__builtin_amdgcn_s_wait_asynccnt)
    __builtin_amdgcn_s_wait_asynccnt(N);
#else
    asm volatile("s_wait_asynccnt %0" :: "i"(N) : "memory");
#endif
}

// ---------------------------------------------------------------------------
// Pass 1: degree histogram of src indices.  3.2M int atomics onto a 400 KB
// L2-resident counter array.  Compiler loads only the low dword of each
// int64 index (values < 2^31), so this streams 12.8 MB + atomics.
// src stays RT-cached: pass 2 re-reads it from L2.
// ---------------------------------------------------------------------------
__global__ void mp_degree_kernel(const long long* __restrict__ src,
                                 int* __restrict__ deg, int E) {
    int e = blockIdx.x * blockDim.x + threadIdx.x;
    if (e < E) {
        int s = (int)src[e];
        atomicAdd(&deg[s], 1);
    }
}

// ---------------------------------------------------------------------------
// Pass 2: out[e,:] = (deg[src[e]] - 1) * x[src[e],:] + extra[src[e],:]
//
//   Two 64-edge tiles per block, double-buffered index staging:
//     issue async(tile0), async(tile1)            -> ASYNCcnt = 2 (waves 0,1)
//     s_wait_asynccnt 1  (loads retire in order)  -> tile0 resident
//     compute tile0 while tile1 is in flight
//     s_wait_asynccnt 0                           -> tile1 resident
//     compute tile1
//
//   4 lanes/edge, one float4 each: a wave32 emits 8 coalesced 128-bit
//   non-temporal stores (th:TH_STORE_NT) covering 512 B of contiguous output,
//   keeping the 205 MB result stream from evicting the L2-resident
//   x/extra/deg/src working set (~39 MB << 192 MB L2).
// ---------------------------------------------------------------------------
__device__ __forceinline__ void mp_process_tile(const float* __restrict__ x,
                                                const float* __restrict__ extra,
                                                const int* __restrict__ deg,
                                                float* __restrict__ out,
                                                const long long* s_src,
                                                int tile_base, int tid, int E) {
    const int le = tid >> 2;           // local edge id   0..63
    const int c  = tid & 3;            // float4 chunk    0..3
    const int e  = tile_base + le;
    if (e >= E) return;

    const int   s     = (int)s_src[le];          // low dword from LDS
    const float scale = (float)(deg[s] - 1);

    const v4f* xrow = (const v4f*)(x     + (size_t)s * FEAT);
    const v4f* erow = (const v4f*)(extra + (size_t)s * FEAT);
    const v4f xv = xrow[c];            // 128-bit gather, L2-resident
    const v4f ev = erow[c];

    v4f o;
    o.x = fmaf(scale, xv.x, ev.x);
    o.y = fmaf(scale, xv.y, ev.y);
    o.z = fmaf(scale, xv.z, ev.z);
    o.w = fmaf(scale, xv.w, ev.w);

    v4f* orow = (v4f*)(out + (size_t)e * FEAT);
    __builtin_nontemporal_store(o, orow + c);    // global_store_b128 ... NT
}

__global__ void mp_apply_kernel(const float* __restrict__ x,
                                const float* __restrict__ extra,
                                const long long* __restrict__ src,
                                const int* __restrict__ deg,
                                float* __restrict__ out, int E) {
    __shared__ long long s_src[2][TILE_EDGES];

    const int tid   = threadIdx.x;
    const int base0 = blockIdx.x * (2 * TILE_EDGES);
    const int base1 = base0 + TILE_EDGES;
    const bool have1 = (base1 < E);    // second tile (at least partially) live

    if (tid < TILE_EDGES) {
        const int e0 = base0 + tid;
        if (e0 < E)          async_copy_b64(src + e0, &s_src[0][tid]);
        const int e1 = base1 + tid;
        if (have1 && e1 < E) async_copy_b64(src + e1, &s_src[1][tid]);
    }

    // Tile 0 ready when at most one async load (tile 1's) is still in flight.
    // If no tile-1 fetch was issued, fall back to a full drain so the
    // partial-count wait can never under-wait.
    if (have1) wait_asynccnt<1>();
    else       wait_asynccnt<0>();
    __syncthreads();

    mp_process_tile(x, extra, deg, out, s_src[0], base0, tid, E);

    if (have1) {
        wait_asynccnt<0>();
        __syncthreads();
        mp_process_tile(x, extra, deg, out, s_src[1], base1, tid, E);
    }
}

// ---------------------------------------------------------------------------
extern "C" void kernel_launch(void* const* d_in, const int* in_sizes, int n_in,
                              void* d_out, int out_size, void* d_ws, size_t ws_size,
                              hipStream_t stream) {
    const float*     x     = (const float*)d_in[0];
    const float*     extra = (const float*)d_in[1];
    const long long* edge  = (const long long*)d_in[2];  // [2, E]; row 0 = src

    const int N = in_sizes[0] / FEAT;   // 100000
    const int E = in_sizes[2] / 2;      // 3200000

    int* deg = (int*)d_ws;              // 4*N = 400 KB scratch

    // must re-zero every call: kernel_launch is replayed via graph capture
    (void)hipMemsetAsync(deg, 0, (size_t)N * sizeof(int), stream);

    {
        dim3 blk(256), grd((E + 255) / 256);
        mp_degree_kernel<<<grd, blk, 0, stream>>>(edge, deg, E);
    }
    {
        const int edges_per_block = 2 * TILE_EDGES;
        dim3 blk(256), grd((E + edges_per_block - 1) / edges_per_block);
        mp_apply_kernel<<<grd, blk, 0, stream>>>(x, extra, edge, deg,
                                                 (float*)d_out, E);
    }
}